// Repro_11879879543049
// MI455X (gfx1250) — compile-verified
//
#include <hip/hip_runtime.h>
#include <stdint.h>

// Problem constants from the reference:
//   cache : (4, 32, 8192, 128) fp32  = 134,217,728 floats = 512 MB
//   update: (4, 32,   16, 128) fp32  = 262,144 floats     = 1 MB
//   pos   : scalar int (0 .. 8192-16)
// Output = cache with update written at seq offset pos.
//
// Pure HBM-bound: ~1.07 GB of traffic -> ~46 us floor at 23.3 TB/s.
// Bulk stream uses the CDNA5 async global<->LDS DMA path (ASYNCcnt):
//  - GVS addressing (SGPR base + 32-bit per-lane offset; tensor < 2 GB)
//  - non-temporal TH hints (1 GB streamed once through a 192 MB L2)

#define TPB     256   // threads per block (8 waves of 32)
#define CHUNKS  8     // 16-byte chunks per thread -> 128 B/thread, 32 KB/block

struct alignas(16) F4 { float x, y, z, w; };

// ---------------------------------------------------------------------------
// Kernel 1: bulk 512 MB copy via async global->LDS->global DMA.
// Each block moves a contiguous 32 KB region. Per iteration k, lane L handles
// chunk (k*TPB + L): consecutive lanes hit consecutive 16 B chunks (coalesced).
// ---------------------------------------------------------------------------
__global__ void __launch_bounds__(TPB)
bulk_copy_async_kernel(const float* __restrict__ src, float* __restrict__ dst) {
    __shared__ alignas(16) char tile[TPB * CHUNKS * 16];   // 32 KB

    // Low 32 bits of a flat shared pointer == LDS byte offset;
    // that is what the async ops' LDS VGPR expects.
    const unsigned lds_base = (unsigned)(uintptr_t)&tile[0];

    // Byte offset of this block's 32 KB region; total tensor is 2^29 bytes,
    // so 32-bit offsets cover everything (GVS: SGPR base + 32-bit lane offset).
    const unsigned block_byte0 = (unsigned)blockIdx.x * (TPB * CHUNKS * 16);
    const unsigned lane16      = threadIdx.x * 16u;

    // Issue 8 async loads (ASYNCcnt++ each); no VGPR data movement.
    // NT hint: 512 MB read exactly once -> don't churn L2.
    #pragma unroll
    for (int k = 0; k < CHUNKS; ++k) {
        const unsigned goff = block_byte0 + (unsigned)(k * TPB * 16) + lane16;
        const unsigned loff = lds_base + (unsigned)(k * TPB * 16) + lane16;
        asm volatile("global_load_async_to_lds_b128 %0, %1, %2 th:TH_LOAD_NT"
                     :: "v"(loff), "v"(goff), "s"(src)
                     : "memory");
    }

    // Wave-private dependency: our stores read only LDS our own loads wrote.
    asm volatile("s_wait_asynccnt 0" ::: "memory");

    // Issue 8 async stores LDS -> global; drained by s_endpgm's implicit
    // wait-idle (all counters, including ASYNCcnt, must reach zero).
    // NT hint: written once, never re-read by this dispatch.
    #pragma unroll
    for (int k = 0; k < CHUNKS; ++k) {
        const unsigned goff = block_byte0 + (unsigned)(k * TPB * 16) + lane16;
        const unsigned loff = lds_base + (unsigned)(k * TPB * 16) + lane16;
        asm volatile("global_store_async_from_lds_b128 %0, %1, %2 th:TH_STORE_NT"
                     :: "v"(goff), "v"(loff), "s"(dst)
                     : "memory");
    }
}

// ---------------------------------------------------------------------------
// Kernel 2: overwrite the 16-row slice at seq offset pos with `update`.
// update float4 grid: (b*h = 128, s = 16, d4 = 32)  -> 65,536 float4s.
// out    float4 row stride per (b,h): 8192*32; per s: 32.
// ---------------------------------------------------------------------------
__global__ void __launch_bounds__(TPB)
write_update_kernel(const float* __restrict__ upd, const int* __restrict__ pos,
                    float* __restrict__ out) {
    const int f = blockIdx.x * TPB + threadIdx.x;   // float4 index in update
    const int d4   = f & 31;          // 0..31   (128 floats / 4)
    const int sidx = (f >> 5) & 15;   // 0..15
    const int bh   = f >> 9;          // 0..127  (4*32)

    // pos fits in the low 32-bit word for either int32 or LE int64 storage.
    const long long p = (long long)pos[0];

    const F4* u = (const F4*)upd;
    F4*       o = (F4*)out;
    const long long oidx = (long long)bh * (8192LL * 32) + (p + sidx) * 32 + d4;
    o[oidx] = u[f];
}

// ---------------------------------------------------------------------------
extern "C" void kernel_launch(void* const* d_in, const int* in_sizes, int n_in,
                              void* d_out, int out_size, void* d_ws, size_t ws_size,
                              hipStream_t stream) {
    const float* cache  = (const float*)d_in[0];
    const float* update = (const float*)d_in[1];
    const int*   pos    = (const int*)d_in[2];
    float*       out    = (float*)d_out;

    // 134,217,728 floats -> 33,554,432 16B chunks -> 16,384 blocks of 2048.
    const long long total_chunks = 134217728LL / 4;
    const int bulk_blocks = (int)(total_chunks / (TPB * CHUNKS));   // 16384
    bulk_copy_async_kernel<<<bulk_blocks, TPB, 0, stream>>>(cache, out);

    // 262,144 floats -> 65,536 float4s -> 256 blocks. Runs after the copy
    // (same stream), overwriting the slice at pos.
    const int upd_f4 = 262144 / 4;
    write_update_kernel<<<upd_f4 / TPB, TPB, 0, stream>>>(update, pos, out);
}